// TrinaryPtimitivesToyotaJoint_45938970198068
// MI455X (gfx1250) — compile-verified
//
#include <hip/hip_runtime.h>
#include <hip/hip_bf16.h>

typedef __attribute__((ext_vector_type(2))) float v2f;
typedef __attribute__((ext_vector_type(8))) float v8f;

#define N_JOINTS  13
#define N_TRIP    2197      // 13^3
#define CMP_DIM   10
#define EPS_SQ    1e-10f

// ADJ[a] bit b == 1  iff edge (a,b) present.
// Edges: (0,2)(2,4)(1,3)(3,5)(6,8)(8,10)(7,9)(9,11)(4,5)(4,10)(5,11)(10,11)(11,12)(10,12)
__device__ __constant__ unsigned int ADJ[N_JOINTS] = {
    0x0004u,  // 0 -> 2
    0x0008u,  // 1 -> 3
    0x0010u,  // 2 -> 4
    0x0020u,  // 3 -> 5
    0x0420u,  // 4 -> 5,10
    0x0800u,  // 5 -> 11
    0x0100u,  // 6 -> 8
    0x0200u,  // 7 -> 9
    0x0400u,  // 8 -> 10
    0x0800u,  // 9 -> 11
    0x1800u,  // 10 -> 11,12
    0x1000u,  // 11 -> 12
    0x0000u   // 12
};

__global__ __launch_bounds__(256)
void triplet_angle_sigmoid_kernel(const float* __restrict__ states,
                                  const float* __restrict__ theta,
                                  const float* __restrict__ gamma,
                                  float* __restrict__ out)
{
    __shared__ float gram[16 * 16];   // padded 13x13 Gram matrix of this frame

    const int frame = blockIdx.x;
    const int tid   = threadIdx.x;

    // ---- Wave 0: Gram matrix G = P * P^T via V_WMMA_F32_16X16X4_F32 ----
    // P is 13x3, zero-padded to 16x4.  For a symmetric product the A fragment
    // (16x4, MxK) and B fragment (4x16, KxN with B[k][n]=P[n][k]) hold
    // identical per-lane register images, so we build one fragment.
    if (tid < 32) {
        const float* P = states + (size_t)frame * (N_JOINTS * 3);
        const int m   = tid & 15;   // matrix row (M) / col (N)
        const int khi = tid >> 4;   // 0: K=0,1   1: K=2,3
        float a0 = 0.0f, a1 = 0.0f;
        if (m < N_JOINTS) {
            a0 = P[m * 3 + khi * 2];          // K = 0 or 2
            if (!khi) a1 = P[m * 3 + 1];      // K = 1 (K=3 stays zero pad)
        }
        v2f A; A.x = a0; A.y = a1;
        v8f acc = {};
        // (neg_a, A, neg_b, B, c_mod, C, reuse_a, reuse_b)
        acc = __builtin_amdgcn_wmma_f32_16x16x4_f32(
            false, A, false, A, (short)0, acc, false, false);

        // D layout: VGPR r -> M=r (lanes 0-15) / M=8+r (lanes 16-31), N = m
        const int rowBase = khi ? 8 : 0;
#pragma unroll
        for (int r = 0; r < 8; ++r)
            gram[(rowBase + r) * 16 + m] = acc[r];
    }
    __syncthreads();

    // ---- All 8 waves: stream 2197 triplets, 10 NT float writes each ----
    float th[CMP_DIM], gm[CMP_DIM];
#pragma unroll
    for (int t = 0; t < CMP_DIM; ++t) { th[t] = theta[t]; gm[t] = gamma[t]; }

    float* obase = out + (size_t)frame * N_TRIP * CMP_DIM;

    for (int i = tid; i < N_TRIP; i += 256) {
        const unsigned a   = (unsigned)i / 169u;
        const unsigned rem = (unsigned)i - a * 169u;
        const unsigned b   = rem / 13u;
        const unsigned c   = rem - b * 13u;

        const float Gab = gram[a * 16 + b];
        const float Gbc = gram[b * 16 + c];
        const float Gac = gram[a * 16 + c];
        const float Gaa = gram[a * 16 + a];
        const float Gbb = gram[b * 16 + b];
        const float Gcc = gram[c * 16 + c];

        const float inner = Gac - Gab - Gbc + Gbb;
        const float n1sq  = Gaa - 2.0f * Gab + Gbb;
        const float n2sq  = Gcc - 2.0f * Gbc + Gbb;
        const float denom = sqrtf(fmaxf(n1sq * n2sq, EPS_SQ));
        const float cosv  = fminf(fmaxf(inner / denom, -0.999f), 0.999f);

        const unsigned msk = ((ADJ[a] >> b) & (ADJ[b] >> c)) & 1u;
        float ang = 0.0f;
        if (msk) ang = acosf(cosv);   // uniformly skipped for ~99% of triplets

        v2f* op = (v2f*)(obase + (size_t)i * CMP_DIM);  // 40B stride -> 8B aligned
#pragma unroll
        for (int t = 0; t < CMP_DIM; t += 2) {
            const float x0 = (ang - th[t])     * gm[t];
            const float x1 = (ang - th[t + 1]) * gm[t + 1];
            v2f v;
            v.x = 1.0f / (1.0f + __expf(-x0));
            v.y = 1.0f / (1.0f + __expf(-x1));
            __builtin_nontemporal_store(v, op + (t >> 1));  // NT: don't thrash L2
        }
    }
}

extern "C" void kernel_launch(void* const* d_in, const int* in_sizes, int n_in,
                              void* d_out, int out_size, void* d_ws, size_t ws_size,
                              hipStream_t stream) {
    (void)n_in; (void)out_size; (void)d_ws; (void)ws_size;
    const float* states = (const float*)d_in[0];
    const float* theta  = (const float*)d_in[1];
    const float* gamma  = (const float*)d_in[2];
    float*       out    = (float*)d_out;

    const int frames = in_sizes[0] / (N_JOINTS * 3);   // 16*128 = 2048
    triplet_angle_sigmoid_kernel<<<dim3(frames), dim3(256), 0, stream>>>(
        states, theta, gamma, out);
}